// GCNNet_34479997452472
// MI455X (gfx1250) — compile-verified
//
#include <hip/hip_runtime.h>
#include <cmath>

// Problem constants (match reference setup_inputs)
static constexpr int B  = 2;
static constexpr int N  = 50000;
static constexpr int E  = 1600000;
static constexpr int C  = 32;
static constexpr int M  = B * N;          // 100000 GEMM rows
static constexpr int MT = M / 16;         // 6250 16-row tiles (exact)

typedef __attribute__((ext_vector_type(2))) float v2f;
typedef __attribute__((ext_vector_type(8))) float v8f;

// ---------------------------------------------------------------------------
// Per-node tanh precompute: t_nid[n] = tanh(node_emb[cent_n_id[n]]),
//                           t_ntp[n] = tanh(node_type_emb[node_type[n]])
// ---------------------------------------------------------------------------
__global__ void node_tanh_kernel(const int* __restrict__ cent,
                                 const int* __restrict__ ntype,
                                 const float* __restrict__ node_emb,
                                 const float* __restrict__ ntype_emb,
                                 float* __restrict__ t_nid,
                                 float* __restrict__ t_ntp)
{
    int i = blockIdx.x * blockDim.x + threadIdx.x;
    if (i < N) {
        t_nid[i] = tanhf(node_emb[cent[i]]);
        t_ntp[i] = tanhf(ntype_emb[ntype[i]]);
    }
}

// ---------------------------------------------------------------------------
// Per-edge scalar factor (identical for both layers; computed once)
// ---------------------------------------------------------------------------
__global__ void efac_kernel(const float* __restrict__ edge_attr,
                            const int* __restrict__ etype,
                            const float* __restrict__ etype_emb,
                            const int* __restrict__ src,
                            const int* __restrict__ dst,
                            const float* __restrict__ t_nid,
                            const float* __restrict__ t_ntp,
                            float* __restrict__ efac)
{
    int e = blockIdx.x * blockDim.x + threadIdx.x;
    if (e < E) {
        float etp = tanhf(etype_emb[etype[e]]);
        int s = src[e], d = dst[e];
        efac[e] = edge_attr[e]
                * (1.0f + etp)
                * (1.0f + t_nid[d] * t_nid[s])
                * (1.0f + t_ntp[d] * t_ntp[s]);
    }
}

// ---------------------------------------------------------------------------
// h = X * W^T + bias via V_WMMA_F32_16X16X4_F32.
// X: [M,32] row-major, W: [32,32] row-major (h[.,d] = sum_c X[.,c]*W[d,c]).
// Each wave computes a 16x32 output tile; 8 K-steps x 2 col-tiles = 16 WMMA.
// B-fragments (whole 4KB W) preloaded per wave and reused across M tiles.
// Dual-store into H (gather source) and ACC (aggregation seed = h).
// ---------------------------------------------------------------------------
__global__ void gemm_wmma_kernel(const float* __restrict__ X,
                                 const float* __restrict__ W,
                                 const float* __restrict__ bias,
                                 float* __restrict__ H,
                                 float* __restrict__ ACC)
{
    const int lane   = threadIdx.x & 31;
    const int wave   = (blockIdx.x * blockDim.x + threadIdx.x) >> 5;
    const int nwaves = (gridDim.x * blockDim.x) >> 5;
    const int half   = lane >> 4;        // 0: K rows 0/1, 1: K rows 2/3
    const int l16    = lane & 15;
    const int kbase  = half << 1;

    // Preload B fragments: B[k][n] = W[n*32 + k]; lane holds col n = t*16+l16.
    v2f bfrag[2][8];
#pragma unroll
    for (int t = 0; t < 2; ++t) {
        const float* wp = W + (t * 16 + l16) * 32;
#pragma unroll
        for (int ks = 0; ks < 8; ++ks)
            bfrag[t][ks] = *(const v2f*)(wp + ks * 4 + kbase);
    }
    const float bv0 = bias[l16];
    const float bv1 = bias[16 + l16];

    for (int tile = wave; tile < MT; tile += nwaves) {
        // A fragment rows: lanes L and L+16 share row m = tile*16 + l16.
        const float* xrow = X + (tile * 16 + l16) * 32;
        v8f c0 = {};
        v8f c1 = {};
#pragma unroll
        for (int ks = 0; ks < 8; ++ks) {
            v2f a = *(const v2f*)(xrow + ks * 4 + kbase);
            c0 = __builtin_amdgcn_wmma_f32_16x16x4_f32(
                     false, a, false, bfrag[0][ks], (short)0, c0, false, false);
            c1 = __builtin_amdgcn_wmma_f32_16x16x4_f32(
                     false, a, false, bfrag[1][ks], (short)0, c1, false, false);
        }
        // D layout: VGPR r -> row r + 8*half, col l16 (c0: n=l16, c1: n=16+l16)
        float* hb = H   + tile * 16 * 32;
        float* ab = ACC + tile * 16 * 32;
#pragma unroll
        for (int r = 0; r < 8; ++r) {
            int row = r + 8 * half;
            float v0 = c0[r] + bv0;
            float v1 = c1[r] + bv1;
            hb[row * 32 + l16]      = v0;
            hb[row * 32 + 16 + l16] = v1;
            ab[row * 32 + l16]      = v0;
            ab[row * 32 + 16 + l16] = v1;
        }
    }
}

// ---------------------------------------------------------------------------
// Edge message + scatter-add: ACC[b,dst,c] += H[b,src,c] * efac[e].
// Thread = (e,b,c) with c fastest -> each wave handles one (e,b):
// uniform src/dst/efac loads, 128B-coalesced gather and 128B atomic burst.
// All tensors are L2-resident (each ~12.8MB << 192MB L2).
// ---------------------------------------------------------------------------
__global__ void edge_scatter_kernel(const float* __restrict__ H,
                                    const int* __restrict__ src,
                                    const int* __restrict__ dst,
                                    const float* __restrict__ efac,
                                    float* __restrict__ ACC)
{
    long long g = (long long)blockIdx.x * blockDim.x + threadIdx.x;
    // total = E*B*32 = gridDim*blockDim exactly (102,400,000)
    int  c = (int)(g & 31);
    int  b = (int)((g >> 5) & 1);
    long long e = g >> 6;
    int  s = src[e];
    int  d = dst[e];
    float f = efac[e];
    float v = H[((long long)b * N + s) * 32 + c] * f;
    // hardware f32 atomic (global_atomic_add_f32, no return)
    unsafeAtomicAdd(&ACC[((long long)b * N + d) * 32 + c], v);
}

// ---------------------------------------------------------------------------
// x_next = leaky_relu(h + aggr) = leaky_relu(ACC)   (ACC was seeded with h)
// ---------------------------------------------------------------------------
__global__ void leaky_kernel(const float* __restrict__ ACC,
                             float* __restrict__ OUT)
{
    int g = blockIdx.x * blockDim.x + threadIdx.x;
    if (g < M * C) {
        float v = ACC[g];
        OUT[g] = (v > 0.0f) ? v : 0.01f * v;
    }
}

// ---------------------------------------------------------------------------
extern "C" void kernel_launch(void* const* d_in, const int* in_sizes, int n_in,
                              void* d_out, int out_size, void* d_ws, size_t ws_size,
                              hipStream_t stream)
{
    const float* x         = (const float*)d_in[0];   // [B,N,C]
    const int*   edge_idx  = (const int*)  d_in[1];   // [2,E]
    const float* edge_attr = (const float*)d_in[2];   // [E]
    const int*   cent_n_id = (const int*)  d_in[3];   // [N]
    const int*   node_type = (const int*)  d_in[4];   // [N]
    const int*   edge_type = (const int*)  d_in[5];   // [E]
    const float* node_emb  = (const float*)d_in[6];   // [N,1]
    const float* ntype_emb = (const float*)d_in[7];   // [4,1]
    const float* etype_emb = (const float*)d_in[8];   // [4,1]
    const float* W1        = (const float*)d_in[9];   // [C,C]
    const float* b1        = (const float*)d_in[10];  // [C]
    const float* W2        = (const float*)d_in[11];  // [C,C]
    const float* b2        = (const float*)d_in[12];  // [C]
    float*       out       = (float*)d_out;           // [B,N,C]

    const int* src = edge_idx;       // edge_index[0]
    const int* dst = edge_idx + E;   // edge_index[1]

    // Workspace layout (floats)
    float* ws    = (float*)d_ws;
    float* t_nid = ws;                       // N
    float* t_ntp = t_nid + N;                // N
    float* efac  = t_ntp + N;                // E
    float* H     = efac + E;                 // M*C
    float* ACC   = H + (size_t)M * C;        // M*C
    float* X1    = ACC + (size_t)M * C;      // M*C  (layer-1 output)

    const int TB = 256;

    // 1) per-node tanh
    node_tanh_kernel<<<(N + TB - 1) / TB, TB, 0, stream>>>(
        cent_n_id, node_type, node_emb, ntype_emb, t_nid, t_ntp);

    // 2) per-edge factor (shared by both layers)
    efac_kernel<<<(E + TB - 1) / TB, TB, 0, stream>>>(
        edge_attr, edge_type, etype_emb, src, dst, t_nid, t_ntp, efac);

    const long long scatter_total  = (long long)E * B * C;   // 102,400,000
    const int       scatter_blocks = (int)(scatter_total / TB);
    const int       elem_blocks    = (M * C + TB - 1) / TB;

    // ---- Layer 1 ----
    gemm_wmma_kernel<<<256, TB, 0, stream>>>(x, W1, b1, H, ACC);
    edge_scatter_kernel<<<scatter_blocks, TB, 0, stream>>>(H, src, dst, efac, ACC);
    leaky_kernel<<<elem_blocks, TB, 0, stream>>>(ACC, X1);

    // ---- Layer 2 ----
    gemm_wmma_kernel<<<256, TB, 0, stream>>>(X1, W2, b2, H, ACC);
    edge_scatter_kernel<<<scatter_blocks, TB, 0, stream>>>(H, src, dst, efac, ACC);
    leaky_kernel<<<elem_blocks, TB, 0, stream>>>(ACC, out);
}